// WalshHadamardCS_12025908429235
// MI455X (gfx1250) — compile-verified
//
#include <hip/hip_runtime.h>
#include <cstdint>

// FWHT (N = 65536 = 16^4) as 4 radix-16 stages of H16 matrix multiplies on the
// CDNA5 matrix core (V_WMMA_F32_16X16X4_F32, exact fp32), with the global->LDS
// row load performed by the Tensor Data Mover (TENSOR_LOAD_TO_LDS, TENSORcnt).
// One workgroup per (batch, channel) row; whole row lives in 256KB of the
// WGP's 320KB LDS. Final pass fuses perm-gather + singulars + 1/img_dim.

typedef __attribute__((ext_vector_type(2))) float        v2f;
typedef __attribute__((ext_vector_type(8))) float        v8f;
typedef __attribute__((ext_vector_type(4))) unsigned int v4u;
typedef __attribute__((ext_vector_type(8))) int          v8i;
typedef __attribute__((ext_vector_type(4))) int          v4i;

#define B_DIM   32
#define C_DIM   3
#define N_DIM   65536      // 256 * 256
#define K_OUT   49152      // C*N/RATIO
#define J_MAX   16384      // K_OUT / C_DIM
#define THREADS 256        // 8 waves (wave32)

extern "C" __global__ __launch_bounds__(THREADS, 1)
void fwht_wmma_kernel(const float* __restrict__ vec,
                      const float* __restrict__ singulars,
                      const int*   __restrict__ perm,
                      float*       __restrict__ out)
{
    extern __shared__ float smem[];    // N_DIM floats = 256 KB dynamic LDS

    const int bc   = blockIdx.x;
    const int b    = bc / C_DIM;
    const int c    = bc % C_DIM;
    const int tid  = threadIdx.x;
    const int lane = tid & 31;
    const int wave = tid >> 5;
    const int half = lane >> 4;        // 0: lanes 0-15, 1: lanes 16-31
    const int l    = lane & 15;

    const float* src = vec + (size_t)b * (C_DIM * N_DIM) + (size_t)c * N_DIM;

    // ---- TDM: DMA one 256x256 fp32 tile (the whole row) global -> LDS ----
    // D# per CDNA5 ISA; 2D tensor, groups 2/3 unused (zero).
    if (wave == 0) {
        const uint64_t ga = (uint64_t)(uintptr_t)src;

        v4u g0;
        g0[0] = 1u;                                   // count=1 valid user D#
        g0[1] = 0u;                                   // lds_addr = 0 (dyn LDS base)
        g0[2] = (uint32_t)ga;                         // global_addr[31:0]
        g0[3] = (uint32_t)((ga >> 32) & 0x01FFFFFFu)  // global_addr[56:32]
              | 0x80000000u;                          // type = 2 ("image")

        v8i g1;
        g1[0] = 0x00020000;  // wg_mask=0 (not in cluster), data_size=2 -> 4B
        g1[1] = 0x01000000;  // tensor_dim0 = 256   (bits[63:48])
        g1[2] = 0x01000000;  // tensor_dim1 = 256   (bits[95:80])
        g1[3] = 0x01000000;  // tile_dim0   = 256   (bits[127:112])
        g1[4] = 0x00000100;  // tile_dim1   = 256, tile_dim2 = 0
        g1[5] = 0x00000100;  // tensor_dim0_stride = 256 (low 32 of 48)
        g1[6] = 0;           // stride hi, tensor_dim1_stride lo
        g1[7] = 0;           // tensor_dim1_stride hi

        const v4i gz4 = {0, 0, 0, 0};
        const v8i gz8 = {0, 0, 0, 0, 0, 0, 0, 0};
        // 6-arg form (amdgpu-toolchain clang-23 / therock-10.0 headers)
        __builtin_amdgcn_tensor_load_to_lds(g0, g1, gz4, gz4, gz8, 0);
        __builtin_amdgcn_s_wait_tensorcnt(0);         // drain TENSORcnt
    }
    __syncthreads();                                  // release all 8 waves

    // ---- A fragments: H16 columns, H[i][j] = (-1)^popc(i&j), per ISA
    // 32-bit A 16x4 layout: lane<16 holds (M=l, K=0/1), lane>=16 (M=l, K=2/3)
    v2f afrag[4];
#pragma unroll
    for (int k = 0; k < 4; ++k) {
        const int j0 = 4 * k + 2 * half;
        afrag[k].x = (__popc(l & j0)       & 1) ? -1.0f : 1.0f;
        afrag[k].y = (__popc(l & (j0 + 1)) & 1) ? -1.0f : 1.0f;
    }

    // ---- 4 radix-16 stages: apply H16 along digit t (stride 16^t) ----
#pragma unroll
    for (int t = 0; t < 4; ++t) {
        const int st = 4 * t;
        const int s  = 1 << st;                 // transform stride
        for (int tile = wave; tile < 256; tile += 8) {
            // vector index for this lane's column of the 16x16 tile
            const int g    = tile * 16 + l;
            const int lo   = g & (s - 1);
            const int hi   = g >> st;
            const int base = (hi << (st + 4)) + lo;

            v8f acc = {};
#pragma unroll
            for (int k = 0; k < 4; ++k) {
                // B 4x16 chunk: lane<16 rows 4k+0/4k+1, lane>=16 rows 4k+2/4k+3
                const int r0 = 4 * k + 2 * half;
                v2f bfrag;
                bfrag.x = smem[base + r0 * s];
                bfrag.y = smem[base + (r0 + 1) * s];
                // D = A(16x4) * B(4x16) + C, exact fp32
                acc = __builtin_amdgcn_wmma_f32_16x16x4_f32(
                        false, afrag[k], false, bfrag,
                        (short)0, acc, false, false);
            }
            // D 16x16 layout: VGPR r holds (M = r + 8*half, N = l)
#pragma unroll
            for (int r = 0; r < 8; ++r) {
                smem[base + (r + 8 * half) * s] = acc[r];
            }
        }
        __syncthreads();
    }

    // ---- Fused permutation gather + singulars + 1/img_dim scale ----
    const float inv = 1.0f / 256.0f;
    float* orow = out + (size_t)b * K_OUT;
    for (int j = tid; j < J_MAX; j += THREADS) {
        const int p = perm[j];
        const int k = j * C_DIM + c;
        orow[k] = singulars[k] * smem[p] * inv;
    }
}

extern "C" void kernel_launch(void* const* d_in, const int* in_sizes, int n_in,
                              void* d_out, int out_size, void* d_ws, size_t ws_size,
                              hipStream_t stream) {
    (void)in_sizes; (void)n_in; (void)out_size; (void)d_ws; (void)ws_size;

    const float* vec       = (const float*)d_in[0];
    const float* singulars = (const float*)d_in[1];
    const int*   perm      = (const int*)d_in[2];
    float*       out       = (float*)d_out;

    const size_t shmem = (size_t)N_DIM * sizeof(float);   // 256 KB (<= 320 KB WGP LDS)
    (void)hipFuncSetAttribute((const void*)fwht_wmma_kernel,
                              hipFuncAttributeMaxDynamicSharedMemorySize, (int)shmem);

    hipLaunchKernelGGL(fwht_wmma_kernel,
                       dim3(B_DIM * C_DIM), dim3(THREADS), shmem, stream,
                       vec, singulars, perm, out);
}